// Encoder_40570261078498
// MI455X (gfx1250) — compile-verified
//
#include <hip/hip_runtime.h>
#include <math.h>

// Problem constants (match reference)
#define F     8
#define DM    256
#define DI    512
#define DS    16
#define DC    4
#define DTR   16
#define NB    4
#define LSEQ  512
#define ROWS  (NB * LSEQ)   // 2048

typedef float v2f __attribute__((ext_vector_type(2)));
typedef float v8f __attribute__((ext_vector_type(8)));

__device__ __forceinline__ float silu_f(float v) { return v / (1.0f + __expf(-v)); }

// Async global->LDS copy (gfx1250, ASYNCcnt-tracked). 16 bytes per lane.
__device__ __forceinline__ void async_b128(unsigned lds_off, const float* g) {
    asm volatile("global_load_async_to_lds_b128 %0, %1, off"
                 :: "v"(lds_off), "v"((unsigned long long)(uintptr_t)g)
                 : "memory");
}

// ---------------------------------------------------------------------------
// Stage 1: sin/cos featurization.  u[f][row][dm]:
//   dm = 2*i -> sin(ipt[row,f]*2*pi*i), dm = 2*i+1 -> cos(...)
// ---------------------------------------------------------------------------
__global__ void k_features(const float* __restrict__ ipt, float* __restrict__ u, int total) {
    int t = blockIdx.x * blockDim.x + threadIdx.x;
    if (t >= total) return;
    int i   = t & 127;              // DM/2 = 128 = 2^7
    int row = (t >> 7) & (ROWS - 1);
    int f   = t >> 18;              // ROWS*128 = 2^18
    float ang = ipt[row * F + f] * (6.283185307179586f * (float)i);
    float s, c;
    sincosf(ang, &s, &c);
    float* up = u + ((size_t)f * ROWS + row) * DM + 2 * i;
    up[0] = s;
    up[1] = c;
}

// ---------------------------------------------------------------------------
// Generic WMMA f32 GEMM:  C[f] = A[f] @ W[f]^T   (+ optional bias+softplus)
//   A: M x K row-major (row stride lda), W: N x K row-major, C: M x N.
//   Block = 64x64 C tile, KT = 16, 256 threads (8 waves, 4m x 2n), each wave
//   owns two V_WMMA_F32_16X16X4_F32 accumulators (16x32 of C).
//   Tiles are double-buffered in LDS and filled with async B128 copies
//   (2 async instructions per wave per tile -> s_wait_asynccnt 2 pipelining).
//   Requires M % 64 == 0, K % 16 == 0, lda % 4 == 0 (true at all call sites).
// ---------------------------------------------------------------------------
template<int EPI>
__global__ void k_gemm(const float* __restrict__ A, int aFS, int lda,
                       const float* __restrict__ W, int bFS,
                       float* __restrict__ C, int cFS,
                       int M, int N, int K,
                       const float* __restrict__ bias, int biasFS) {
    // [buf][row][k], 20-float row stride: 16B-aligned rows for B128 writes,
    // conflict-free b64 reads for the WMMA operand pattern.
    __shared__ float lsa[2][64][20];
    __shared__ float lsb[2][64][20];

    const int f = blockIdx.z;
    const float* Af = A + (size_t)f * aFS;
    const float* Wf = W + (size_t)f * bFS;
    float*       Cf = C + (size_t)f * cFS;

    const int mBase = blockIdx.y * 64;
    const int nBase = blockIdx.x * 64;

    // Loader mapping: 4 lanes per row, one B128 each (row stride lda/K floats).
    const int tRow = threadIdx.x >> 2;          // 0..63
    const int tK4  = (threadIdx.x & 3) * 4;     // 0,4,8,12
    const float* gA = Af + (size_t)(mBase + tRow) * lda + tK4;
    int nRow = nBase + tRow;
    if (nRow > N - 1) nRow = N - 1;             // clamp: garbage cols never stored
    const float* gB = Wf + (size_t)nRow * K + tK4;
    const unsigned laA0 = (unsigned)(uintptr_t)&lsa[0][tRow][tK4];
    const unsigned laA1 = (unsigned)(uintptr_t)&lsa[1][tRow][tK4];
    const unsigned laB0 = (unsigned)(uintptr_t)&lsb[0][tRow][tK4];
    const unsigned laB1 = (unsigned)(uintptr_t)&lsb[1][tRow][tK4];

    // WMMA lane mapping (ISA 16x16x4 f32 layout).
    const int lane = threadIdx.x & 31;
    const int ln   = lane & 15;
    const bool hi  = lane >= 16;
    const int wave = threadIdx.x >> 5;
    const int wm   = wave & 3;
    const int wn   = wave >> 2;
    const int am   = wm * 16 + ln;
    const int bn0  = wn * 32 + ln;
    const int kl   = hi ? 2 : 0;

    v8f acc0 = {};
    v8f acc1 = {};

    // Prologue: fill buffer 0.
    async_b128(laA0, gA);
    async_b128(laB0, gB);

    const int nt = K >> 4;
    for (int t = 0; t < nt; t++) {
        const int cur = t & 1;
        if (t + 1 < nt) {
            // Issue next tile into the other buffer, then wait for current
            // tile only (async loads complete in order).
            async_b128(cur ? laA0 : laA1, gA + (t + 1) * 16);
            async_b128(cur ? laB0 : laB1, gB + (t + 1) * 16);
            asm volatile("s_wait_asynccnt 0x2" ::: "memory");
        } else {
            asm volatile("s_wait_asynccnt 0x0" ::: "memory");
        }
        __syncthreads();

        #pragma unroll
        for (int kk = 0; kk < 16; kk += 4) {
            v2f a  = *(const v2f*)&lsa[cur][am][kk + kl];
            v2f b0 = *(const v2f*)&lsb[cur][bn0][kk + kl];
            v2f b1 = *(const v2f*)&lsb[cur][bn0 + 16][kk + kl];
            acc0 = __builtin_amdgcn_wmma_f32_16x16x4_f32(false, a, false, b0,
                                                         (short)0, acc0, false, false);
            acc1 = __builtin_amdgcn_wmma_f32_16x16x4_f32(false, a, false, b1,
                                                         (short)0, acc1, false, false);
        }
        __syncthreads();   // all waves done with buf[cur] before it is refilled
    }

    // Store (C/D layout: row = base + r + 8*(lane>=16), col = lane%16).
    const int rowBase = mBase + wm * 16 + (hi ? 8 : 0);
    const int colA    = nBase + wn * 32 + ln;
    const float* bf = (EPI == 1) ? (bias + (size_t)f * biasFS) : nullptr;
    #pragma unroll
    for (int r = 0; r < 8; r++) {
        int row = rowBase + r;
        float v0 = acc0[r];
        float v1 = acc1[r];
        if (EPI == 1) {   // bias + softplus
            if (colA < N)      { v0 += bf[colA];      v0 = (v0 > 20.f) ? v0 : log1pf(__expf(v0)); }
            if (colA + 16 < N) { v1 += bf[colA + 16]; v1 = (v1 > 20.f) ? v1 : log1pf(__expf(v1)); }
        }
        if (colA < N)      Cf[(size_t)row * N + colA]      = v0;
        if (colA + 16 < N) Cf[(size_t)row * N + colA + 16] = v1;
    }
}

// ---------------------------------------------------------------------------
// Stage 3: depthwise causal conv (DC=4) + bias + SiLU.
// ---------------------------------------------------------------------------
__global__ void k_conv(const float* __restrict__ xz, const float* __restrict__ cw,
                       const float* __restrict__ cb, float* __restrict__ xc, int total) {
    int t = blockIdx.x * blockDim.x + threadIdx.x;
    if (t >= total) return;
    int d   = t & (DI - 1);
    int row = (t >> 9) & (ROWS - 1);    // DI = 2^9
    int f   = t >> 20;                  // ROWS*DI = 2^20
    int l   = row & (LSEQ - 1);
    const float* xp = xz + ((size_t)f * ROWS + row) * (2 * DI) + d;
    const float* w  = cw + ((size_t)f * DI + d) * DC;
    float acc = 0.0f;
    #pragma unroll
    for (int j = 0; j < DC; j++) {
        int dl = j - (DC - 1);          // -3..0
        if (l + dl >= 0) acc += xp[dl * (2 * DI)] * w[j];
    }
    acc += cb[f * DI + d];
    xc[((size_t)f * ROWS + row) * DI + d] = silu_f(acc);
}

// ---------------------------------------------------------------------------
// Stage 6: selective scan.  One block per (f,b); 512 threads (one per d).
//   h[16] + A-row in registers. B_t/C_t addresses are wave-uniform, so the
//   float4 loads broadcast from cache -- no LDS, no barriers.
//   Writes gated output in place over xc.
// ---------------------------------------------------------------------------
__global__ void k_scan(const float* __restrict__ dtbc, const float* __restrict__ dt,
                       float* __restrict__ xc, const float* __restrict__ xz,
                       const float* __restrict__ A_log, const float* __restrict__ Dp) {
    const int f = blockIdx.x >> 2;
    const int b = blockIdx.x & 3;
    const int d = threadIdx.x;

    float Aexp[DS];
    #pragma unroll
    for (int s = 0; s < DS; s++)
        Aexp[s] = -__expf(A_log[((size_t)f * DI + d) * DS + s]);
    const float Dd = Dp[f * DI + d];

    float h[DS];
    #pragma unroll
    for (int s = 0; s < DS; s++) h[s] = 0.0f;

    const size_t fr = (size_t)f * ROWS;
    for (int t = 0; t < LSEQ; t++) {
        const int row = b * LSEQ + t;
        const float4* bp = (const float4*)(dtbc + (fr + row) * 48 + DTR);
        float bc[2 * DS];
        #pragma unroll
        for (int q = 0; q < 8; q++) {
            float4 v = bp[q];
            bc[4 * q + 0] = v.x; bc[4 * q + 1] = v.y;
            bc[4 * q + 2] = v.z; bc[4 * q + 3] = v.w;
        }
        const float dtv = dt[(fr + row) * DI + d];
        const float xv  = xc[(fr + row) * DI + d];
        const float dx  = dtv * xv;
        float y = 0.0f;
        #pragma unroll
        for (int s = 0; s < DS; s++) {
            h[s] = h[s] * __expf(dtv * Aexp[s]) + dx * bc[s];
            y += h[s] * bc[DS + s];
        }
        const float zv = xz[(fr + row) * (2 * DI) + DI + d];
        xc[(fr + row) * DI + d] = (y + xv * Dd) * silu_f(zv);
    }
}

// ---------------------------------------------------------------------------
// Stage 8: max over 8 consecutive dm of one feature.
// ---------------------------------------------------------------------------
__global__ void k_maxout(const float* __restrict__ outf, float* __restrict__ out, int total) {
    int t = blockIdx.x * blockDim.x + threadIdx.x;
    if (t >= total) return;
    int col = t & 255;
    int row = t >> 8;
    int f = col >> 5;
    int j = col & 31;
    const float* p = outf + ((size_t)f * ROWS + row) * DM + j * 8;
    float m = p[0];
    #pragma unroll
    for (int e = 1; e < 8; e++) m = fmaxf(m, p[e]);
    out[t] = m;
}

// ---------------------------------------------------------------------------
extern "C" void kernel_launch(void* const* d_in, const int* in_sizes, int n_in,
                              void* d_out, int out_size, void* d_ws, size_t ws_size,
                              hipStream_t stream) {
    const float* ipt    = (const float*)d_in[0];
    const float* in_w   = (const float*)d_in[1];
    const float* conv_w = (const float*)d_in[2];
    const float* conv_b = (const float*)d_in[3];
    const float* xproj  = (const float*)d_in[4];
    const float* dtp_w  = (const float*)d_in[5];
    const float* dtp_b  = (const float*)d_in[6];
    const float* A_log  = (const float*)d_in[7];
    const float* Dp     = (const float*)d_in[8];
    const float* out_w  = (const float*)d_in[9];
    float* out = (float*)d_out;
    float* ws  = (float*)d_ws;

    // Workspace layout (floats). u and dt share a region (disjoint lifetimes);
    // outf reuses xz (z is dead after the scan).
    float* buf_xz   = ws;                                      // F*ROWS*1024
    float* buf_xc   = buf_xz   + (size_t)F * ROWS * (2 * DI);  // F*ROWS*512
    float* buf_dtbc = buf_xc   + (size_t)F * ROWS * DI;        // F*ROWS*48
    float* buf_r2   = buf_dtbc + (size_t)F * ROWS * 48;        // max(u, dt)
    float* buf_u    = buf_r2;
    float* buf_dt   = buf_r2;
    float* buf_outf = buf_xz;

    // 1. features
    {
        int total = F * ROWS * (DM / 2);
        k_features<<<(total + 255) / 256, 256, 0, stream>>>(ipt, buf_u, total);
    }
    // 2. xz = u @ in_w^T                (M=2048, N=1024, K=256)
    k_gemm<0><<<dim3((2 * DI) / 64, ROWS / 64, F), 256, 0, stream>>>(
        buf_u, ROWS * DM, DM, in_w, (2 * DI) * DM,
        buf_xz, ROWS * (2 * DI), ROWS, 2 * DI, DM, nullptr, 0);
    // 3. depthwise conv + SiLU
    {
        int total = F * ROWS * DI;
        k_conv<<<(total + 255) / 256, 256, 0, stream>>>(buf_xz, conv_w, conv_b, buf_xc, total);
    }
    // 4. dtBC = xc @ xproj^T            (M=2048, N=48, K=512)
    k_gemm<0><<<dim3(1, ROWS / 64, F), 256, 0, stream>>>(
        buf_xc, ROWS * DI, DI, xproj, 48 * DI,
        buf_dtbc, ROWS * 48, ROWS, 48, DI, nullptr, 0);
    // 5. dt = softplus(dtBC[:, :16] @ dtp_w^T + dtp_b)   (M=2048, N=512, K=16)
    k_gemm<1><<<dim3(DI / 64, ROWS / 64, F), 256, 0, stream>>>(
        buf_dtbc, ROWS * 48, 48, dtp_w, DI * DTR,
        buf_dt, ROWS * DI, ROWS, DI, DTR, dtp_b, DI);
    // 6. selective scan + gating (in place over xc)
    k_scan<<<F * NB, DI, 0, stream>>>(buf_dtbc, buf_dt, buf_xc, buf_xz, A_log, Dp);
    // 7. outf = y @ out_w^T             (M=2048, N=256, K=512)
    k_gemm<0><<<dim3(DM / 64, ROWS / 64, F), 256, 0, stream>>>(
        buf_xc, ROWS * DI, DI, out_w, DM * DI,
        buf_outf, ROWS * DM, ROWS, DM, DI, nullptr, 0);
    // 8. max over groups of 8
    {
        int total = ROWS * DM;
        k_maxout<<<(total + 255) / 256, 256, 0, stream>>>(buf_outf, out, total);
    }
}